// EncoderBlock_56435870269921
// MI455X (gfx1250) — compile-verified
//
#include <hip/hip_runtime.h>
#include <math.h>

typedef __attribute__((ext_vector_type(2))) float v2f;
typedef __attribute__((ext_vector_type(8))) float v8f;
typedef __attribute__((ext_vector_type(4))) unsigned int v4u;
typedef __attribute__((ext_vector_type(8))) int v8i;
typedef __attribute__((ext_vector_type(4))) int v4i;

#define LN_EPS 1e-5f
#define INV_SQRT_HS 0.17677669529663687f /* 1/sqrt(32) */

// ---------------------------------------------------------------------------
// fp32 WMMA GEMM: C(MxN) = A(MxK) @ B(KxN), optional fused bias+ReLU.
// Block = 4 waves = one 64x16 output strip (same N-tile). The shared KxN16
// B tile (<=16KB) is staged into LDS once per block by the Tensor Data Mover
// (tensor_load_to_lds, 2D D# descriptor), then each wave runs the K-loop with
// V_WMMA_F32_16X16X4_F32, reading B fragments from LDS.
// Fragment layouts per CDNA5 ISA 7.12.2:
//   A 16x4 : lane l -> M=l%16, K half = (l/16)*2, vgpr j -> K=half+j
//   B 4x16 : lane l -> N=l%16, K half = (l/16)*2, vgpr j -> K=half+j
//   C 16x16: vgpr r -> M = r + (l/16)*8, N = l%16
// N, K multiples of 16; K <= 256 at every call site.
// ---------------------------------------------------------------------------
template <bool BIAS_RELU>
__global__ __launch_bounds__(128) void wmma_gemm_f32(
    const float* __restrict__ A, const float* __restrict__ B,
    const float* __restrict__ bias, float* __restrict__ C,
    int M, int N, int K)
{
  __shared__ float Bs[256 * 16];          // K_max x 16 (16 KB of 320 KB/WGP)

  const int lane   = threadIdx.x & 31;
  const int wv     = threadIdx.x >> 5;    // 0..3
  const int ntn    = N >> 4;
  const int mtiles = M >> 4;
  const int nt     = blockIdx.x % ntn;
  const int mt     = (blockIdx.x / ntn) * 4 + wv;
  const int n0     = nt << 4;
  const bool valid = mt < mtiles;

  // ---- TDM: stage B[0:K, n0:n0+16] -> Bs (row-major K x 16) ----
  if (wv == 0) {
    unsigned long long ga = (unsigned long long)(const void*)(B + n0);
    unsigned int lds = (unsigned int)(size_t)(void*)Bs;
    v4u g0;
    g0[0] = 1u;                                          // count=1, user mode
    g0[1] = lds;                                         // lds_addr
    g0[2] = (unsigned int)(ga & 0xFFFFFFFFull);          // global_addr[31:0]
    g0[3] = (unsigned int)((ga >> 32) & 0x1FFFFFFull)    // global_addr[56:32]
          | (2u << 30);                                  // type=2 ("image")
    v8i g1;
    g1[0] = (2 << 16);                  // wg_mask=0, data_size=2 (4 bytes)
    g1[1] = (N & 0xFFFF) << 16;         // tensor_dim0[15:0] (atomic_bar_addr=0)
    g1[2] = ((K & 0xFFFF) << 16) | ((N >> 16) & 0xFFFF); // dim1.lo | dim0.hi
    g1[3] = (16 << 16) | ((K >> 16) & 0xFFFF);           // tile_dim0=16 | dim1.hi
    g1[4] = K;                          // tile_dim1=K (tile_dim2=0 -> unused)
    g1[5] = N;                          // tensor_dim0_stride[31:0]
    g1[6] = 0;                          // stride0.hi=0, stride1.lo=0
    g1[7] = 0;
    v4i gz4 = {0, 0, 0, 0};             // groups 2/3 unused (2D tile)
    v8i gz8 = {0, 0, 0, 0, 0, 0, 0, 0}; // extra group (clang-23 6-arg form)
    __builtin_amdgcn_tensor_load_to_lds(g0, g1, gz4, gz4, gz8, 0);
    __builtin_amdgcn_s_wait_tensorcnt(0);
  }
  __syncthreads();

  const int am = ((valid ? mt : 0) << 4) + (lane & 15);
  const int kk = (lane >> 4) << 1;      // 0 (lanes 0-15) or 2 (lanes 16-31)
  const int nl = lane & 15;

  const float* __restrict__ arow = A + (size_t)am * K;

  v8f acc = {0.f, 0.f, 0.f, 0.f, 0.f, 0.f, 0.f, 0.f};
#pragma unroll 4
  for (int k = 0; k < K; k += 4) {
    v2f a = *reinterpret_cast<const v2f*>(arow + k + kk);  // 8B aligned (K%4==0)
    v2f b;
    b[0] = Bs[(k + kk + 0) * 16 + nl];
    b[1] = Bs[(k + kk + 1) * 16 + nl];
    __builtin_prefetch(arow + k + 64, 0, 0);               // global_prefetch_b8
    acc = __builtin_amdgcn_wmma_f32_16x16x4_f32(
        /*neg_a=*/false, a, /*neg_b=*/false, b,
        /*c_mod=*/(short)0, acc, /*reuse_a=*/false, /*reuse_b=*/false);
  }

  if (valid) {
    const int rbase = (mt << 4) + ((lane >> 4) << 3);      // +0 or +8
    const int col   = n0 + nl;
#pragma unroll
    for (int r = 0; r < 8; ++r) {
      float v = acc[r];
      if (BIAS_RELU) { v += bias[col]; v = v > 0.f ? v : 0.f; }
      C[(size_t)(rbase + r) * N + col] = v;
    }
  }
}

// ---------------------------------------------------------------------------
// feat[c, l, e] = sff[c, e, l]   (sff flattened (cam, E, L); feat (cam*L, E))
// ---------------------------------------------------------------------------
__global__ __launch_bounds__(256) void transpose_feat(
    const float* __restrict__ sff, float* __restrict__ feat,
    int cam, int L, int E)
{
  int i = blockIdx.x * blockDim.x + threadIdx.x;
  int total = cam * L * E;
  if (i >= total) return;
  int e = i % E;
  int l = (i / E) % L;
  int c = i / (E * L);
  feat[i] = sff[((size_t)c * E + e) * L + l];
}

// ---------------------------------------------------------------------------
// Temporal self-attention + residual + LayerNorm. One wave (32 lanes) per
// query row; lane owns 8 contiguous channels -> a head (32 ch) = 4 lanes.
// ---------------------------------------------------------------------------
__global__ __launch_bounds__(256) void temporal_attn_ln(
    const float* __restrict__ hist, const float* __restrict__ qin,
    const float* __restrict__ vall, const int* __restrict__ idx,
    const float* __restrict__ refw, const float* __restrict__ gamma,
    const float* __restrict__ beta, float* __restrict__ out, int S)
{
  const int lane = threadIdx.x & 31;
  const int s = blockIdx.x * 8 + (threadIdx.x >> 5);
  if (s >= S) return;
  const int e0 = lane << 3;

  const float* qr = qin + (size_t)s * 256;
  float q[8], o[8];
#pragma unroll
  for (int j = 0; j < 8; ++j) q[j] = qr[e0 + j];

  int rows[4];
  float sc[4];
#pragma unroll
  for (int r = 0; r < 4; ++r) {
    rows[r] = idx[s * 4 + r];
    const float* kr = hist + (size_t)rows[r] * 256 + e0;
    float p = 0.f;
#pragma unroll
    for (int j = 0; j < 8; ++j) p += q[j] * kr[j];
    p += __shfl_xor(p, 1);           // reduce across the 4-lane head group
    p += __shfl_xor(p, 2);
    sc[r] = p * INV_SQRT_HS;
  }

  float mx = fmaxf(fmaxf(sc[0], sc[1]), fmaxf(sc[2], sc[3]));
  float w[4], den = 0.f;
#pragma unroll
  for (int r = 0; r < 4; ++r) { w[r] = __expf(sc[r] - mx); den += w[r]; }
  float inv = 1.f / den;
#pragma unroll
  for (int r = 0; r < 4; ++r) w[r] = w[r] * inv * refw[s * 4 + r];

#pragma unroll
  for (int j = 0; j < 8; ++j) o[j] = q[j];               // residual (bev_query)
#pragma unroll
  for (int r = 0; r < 4; ++r) {
    const float* vr = vall + (size_t)rows[r] * 256 + e0;
#pragma unroll
    for (int j = 0; j < 8; ++j) o[j] += w[r] * vr[j];
  }

  float sum = 0.f, sq = 0.f;
#pragma unroll
  for (int j = 0; j < 8; ++j) { sum += o[j]; sq += o[j] * o[j]; }
#pragma unroll
  for (int m = 1; m < 32; m <<= 1) {
    sum += __shfl_xor(sum, m);
    sq  += __shfl_xor(sq,  m);
  }
  float mean = sum * (1.f / 256.f);
  float var  = sq  * (1.f / 256.f) - mean * mean;
  float rstd = rsqrtf(var + LN_EPS);

  float* orow = out + (size_t)s * 256;
#pragma unroll
  for (int j = 0; j < 8; ++j)
    orow[e0 + j] = (o[j] - mean) * rstd * gamma[e0 + j] + beta[e0 + j];
}

// ---------------------------------------------------------------------------
// Spatial cross-attention (cam=6, nr=4 -> 24 joint refs) + residual + LN.
// ---------------------------------------------------------------------------
__global__ __launch_bounds__(256) void spatial_attn_ln(
    const float* __restrict__ feat, const float* __restrict__ vfeat,
    const float* __restrict__ xin, const int* __restrict__ idx,
    const float* __restrict__ refw, const float* __restrict__ gamma,
    const float* __restrict__ beta, float* __restrict__ out, int S, int L)
{
  const int lane = threadIdx.x & 31;
  const int s = blockIdx.x * 8 + (threadIdx.x >> 5);
  if (s >= S) return;
  const int e0 = lane << 3;

  const float* qr = xin + (size_t)s * 256;
  float q[8], o[8];
#pragma unroll
  for (int j = 0; j < 8; ++j) q[j] = qr[e0 + j];

  int rows[24];
  float sc[24];
#pragma unroll
  for (int c = 0; c < 6; ++c) {
#pragma unroll
    for (int r = 0; r < 4; ++r) {
      const int i = c * 4 + r;
      const int l = idx[((size_t)c * S + s) * 4 + r];
      rows[i] = c * L + l;
      const float* kr = feat + (size_t)rows[i] * 256 + e0;
      float p = 0.f;
#pragma unroll
      for (int j = 0; j < 8; ++j) p += q[j] * kr[j];
      p += __shfl_xor(p, 1);
      p += __shfl_xor(p, 2);
      sc[i] = p * INV_SQRT_HS;
    }
  }

  float mx = -3.402823466e38f;
#pragma unroll
  for (int i = 0; i < 24; ++i) mx = fmaxf(mx, sc[i]);
  float den = 0.f;
#pragma unroll
  for (int i = 0; i < 24; ++i) { sc[i] = __expf(sc[i] - mx); den += sc[i]; }
  float inv = 1.f / den;
#pragma unroll
  for (int c = 0; c < 6; ++c)
#pragma unroll
    for (int r = 0; r < 4; ++r)
      sc[c * 4 + r] = sc[c * 4 + r] * inv * refw[((size_t)c * S + s) * 4 + r];

#pragma unroll
  for (int j = 0; j < 8; ++j) o[j] = q[j];               // residual (LN1 output)
#pragma unroll
  for (int i = 0; i < 24; ++i) {
    const float* vr = vfeat + (size_t)rows[i] * 256 + e0;
#pragma unroll
    for (int j = 0; j < 8; ++j) o[j] += sc[i] * vr[j];
  }

  float sum = 0.f, sq = 0.f;
#pragma unroll
  for (int j = 0; j < 8; ++j) { sum += o[j]; sq += o[j] * o[j]; }
#pragma unroll
  for (int m = 1; m < 32; m <<= 1) {
    sum += __shfl_xor(sum, m);
    sq  += __shfl_xor(sq,  m);
  }
  float mean = sum * (1.f / 256.f);
  float var  = sq  * (1.f / 256.f) - mean * mean;
  float rstd = rsqrtf(var + LN_EPS);

  float* orow = out + (size_t)s * 256;
#pragma unroll
  for (int j = 0; j < 8; ++j)
    orow[e0 + j] = (o[j] - mean) * rstd * gamma[e0 + j] + beta[e0 + j];
}

// ---------------------------------------------------------------------------
// out = LayerNorm(y + b2 + xres; g3, be3) — final FFN epilogue.
// ---------------------------------------------------------------------------
__global__ __launch_bounds__(256) void bias_residual_ln(
    const float* __restrict__ y, const float* __restrict__ b2,
    const float* __restrict__ xres, const float* __restrict__ gamma,
    const float* __restrict__ beta, float* __restrict__ out, int S)
{
  const int lane = threadIdx.x & 31;
  const int s = blockIdx.x * 8 + (threadIdx.x >> 5);
  if (s >= S) return;
  const int e0 = lane << 3;

  const float* yr = y + (size_t)s * 256;
  const float* xr = xres + (size_t)s * 256;
  float o[8];
#pragma unroll
  for (int j = 0; j < 8; ++j) o[j] = yr[e0 + j] + b2[e0 + j] + xr[e0 + j];

  float sum = 0.f, sq = 0.f;
#pragma unroll
  for (int j = 0; j < 8; ++j) { sum += o[j]; sq += o[j] * o[j]; }
#pragma unroll
  for (int m = 1; m < 32; m <<= 1) {
    sum += __shfl_xor(sum, m);
    sq  += __shfl_xor(sq,  m);
  }
  float mean = sum * (1.f / 256.f);
  float var  = sq  * (1.f / 256.f) - mean * mean;
  float rstd = rsqrtf(var + LN_EPS);

  float* orow = out + (size_t)s * 256;
#pragma unroll
  for (int j = 0; j < 8; ++j)
    orow[e0 + j] = (o[j] - mean) * rstd * gamma[e0 + j] + beta[e0 + j];
}

// ---------------------------------------------------------------------------
extern "C" void kernel_launch(void* const* d_in, const int* in_sizes, int n_in,
                              void* d_out, int out_size, void* d_ws, size_t ws_size,
                              hipStream_t stream)
{
  (void)n_in; (void)out_size; (void)ws_size;

  const float* history = (const float*)d_in[0];
  const float* query   = (const float*)d_in[1];
  const float* sff     = (const float*)d_in[2];
  const int*   idx_t   = (const int*)  d_in[3];
  const int*   idx_s   = (const int*)  d_in[4];
  const float* ref_w_t = (const float*)d_in[5];
  const float* ref_w_s = (const float*)d_in[6];
  const float* W_v_t   = (const float*)d_in[7];
  const float* W_v_s   = (const float*)d_in[8];
  const float* W1      = (const float*)d_in[9];
  const float* b1      = (const float*)d_in[10];
  const float* W2      = (const float*)d_in[11];
  const float* b2      = (const float*)d_in[12];
  const float* g1      = (const float*)d_in[13];
  const float* be1     = (const float*)d_in[14];
  const float* g2      = (const float*)d_in[15];
  const float* be2     = (const float*)d_in[16];
  const float* g3      = (const float*)d_in[17];
  const float* be3     = (const float*)d_in[18];

  const int E = 256, HID = 128, CAM = 6;
  const int S = in_sizes[1] / E;            // 10000
  const int L = in_sizes[2] / (CAM * E);    // 1024

  // Workspace layout (floats), lifetime-based reuse (~38.2 MB total):
  //   bufA : stage1 v_all (S*E) ; stage2 feat (CAM*L*E) + vfeat (CAM*L*E)
  //   x1   : LN1 output; reused as FFN output y after spatial stage
  //   x2   : LN2 output (residual for final LN)
  //   h    : FFN hidden
  float* ws = (float*)d_ws;
  const size_t featN = (size_t)CAM * L * E;
  const size_t xN    = (size_t)S * E;
  const size_t bufA  = (2 * featN > xN) ? 2 * featN : xN;

  float* v_all = ws;
  float* feat  = ws;
  float* vfeat = ws + featN;
  float* x1    = ws + bufA;
  float* x2    = x1 + xN;
  float* h     = x2 + xN;
  float* y     = x1;                        // x1 dead after spatial stage

  // blocks: 4 M-tiles per block (shared LDS B tile), one N-tile per block
  auto gemm_blocks = [](int M, int N) { return ((M / 16 + 3) / 4) * (N / 16); };

  // 1) v_all = history @ W_v_t            (10000x256x256, fp32 WMMA + TDM)
  wmma_gemm_f32<false><<<gemm_blocks(S, E), 128, 0, stream>>>(
      history, W_v_t, nullptr, v_all, S, E, E);
  // 2) x1 = LN1(temporal_attn + bev_query)
  temporal_attn_ln<<<(S + 7) / 8, 256, 0, stream>>>(
      history, query, v_all, idx_t, ref_w_t, g1, be1, x1, S);
  // 3) feat = transpose(single_frame_feature)
  const int tot = CAM * L * E;
  transpose_feat<<<(tot + 255) / 256, 256, 0, stream>>>(sff, feat, CAM, L, E);
  // 4) vfeat = feat @ W_v_s               (6144x256x256, fp32 WMMA + TDM)
  wmma_gemm_f32<false><<<gemm_blocks(CAM * L, E), 128, 0, stream>>>(
      feat, W_v_s, nullptr, vfeat, CAM * L, E, E);
  // 5) x2 = LN2(spatial_attn + x1)
  spatial_attn_ln<<<(S + 7) / 8, 256, 0, stream>>>(
      feat, vfeat, x1, idx_s, ref_w_s, g2, be2, x2, S, L);
  // 6) h = relu(x2 @ W1 + b1)             (10000x128x256, fused epilogue)
  wmma_gemm_f32<true><<<gemm_blocks(S, HID), 128, 0, stream>>>(
      x2, W1, b1, h, S, HID, E);
  // 7) y = h @ W2                          (10000x256x128)
  wmma_gemm_f32<false><<<gemm_blocks(S, E), 128, 0, stream>>>(
      h, W2, nullptr, y, S, E, HID);
  // 8) out = LN3(y + b2 + x2)
  bias_residual_ln<<<(S + 7) / 8, 256, 0, stream>>>(
      y, b2, x2, g3, be3, (float*)d_out, S);
}